// GraphCNN_46935402611306
// MI455X (gfx1250) — compile-verified
//
#include <hip/hip_runtime.h>
#include <hip/hip_bf16.h>

// GraphCNN for MI455X (gfx1250): fp32 WMMA MLPs + L2-resident atomic scatter.
//
// Sizes (fixed by the reference):
#define N_NODES  131072
#define N_EDGES  2097152
#define MAX_DEPTH 8
#define DIM      64

typedef __attribute__((ext_vector_type(2))) float v2f;
typedef __attribute__((ext_vector_type(8))) float v8f;

// ---- LDS layout (dwords) -------------------------------------------------
// Wt  : one layer's 64x64 weights, transposed (Wt[n][k]), stride 68 to
//       spread ds_load_b64 fragment fetches across banks.
// bias: all 3 layers' biases (192 floats).
// tile: per-wave 16x64 activation tile, stride 68 (wave-private: the
//       C-layout -> A-layout relayout between chained layers goes through
//       here with no workgroup barrier needed).
#define WT_STRIDE   68
#define W_OFF       0                      // 64*68 = 4352 dwords
#define B_OFF       4352                   // 192 dwords
#define TILE_OFF    4544
#define TILE_STRIDE 68                     // 16*68 = 1088 dwords per wave
#define LDS_DWORDS  (TILE_OFF + 8 * 16 * TILE_STRIDE)   // 13248 dw = 52992 B

__device__ __forceinline__ float lrelu(float v) {
    // leaky_relu(x, 0.01) == max(x,0) + 0.01*min(x,0), exact
    return fmaxf(v, 0.0f) + 0.01f * fminf(v, 0.0f);
}

// One fused 3-layer MLP pass over 128 rows per block (8 waves x 16 rows).
// If xres != nullptr: out = xres + act(mlp(in)) * mask   (residual fuse)
// else:               out = act(mlp(in))
__global__ __launch_bounds__(256)
void mlp3_kernel(const float* __restrict__ in,
                 const float* __restrict__ W,     // [3][64][64], (k,n) row-major
                 const float* __restrict__ Bv,    // [3][64]
                 const float* __restrict__ xres,  // [N,64] or nullptr
                 const float* __restrict__ mask,  // [N] or nullptr
                 float* __restrict__ out)
{
    __shared__ float lds[LDS_DWORDS];
    const int tid   = threadIdx.x;
    const int wave  = tid >> 5;
    const int lane  = tid & 31;
    const int node0 = blockIdx.x * 128 + wave * 16;     // wave's 16 rows
    float* tile = &lds[TILE_OFF + wave * (16 * TILE_STRIDE)];

    // biases for all 3 layers (once)
    if (tid < 3 * DIM) lds[B_OFF + tid] = Bv[tid];

    // Stage this wave's 16x64 input tile into LDS, coalesced b128 loads.
    // Wave-private region: same-wave DS ordering makes later ds_loads safe.
    #pragma unroll
    for (int i = 0; i < 8; ++i) {
        int flat = i * 128 + lane * 4;                  // 0..1020
        int r = flat >> 6, c = flat & 63;
        float4 v = *(const float4*)(in + (size_t)(node0 + r) * DIM + c);
        *(float4*)(tile + r * TILE_STRIDE + c) = v;
    }

    // WMMA f32 16x16x4 lane roles (32-bit A 16x4 / B 4x16 layouts):
    //   lanes 0-15 : row/col = lane,      K pair {k, k+1}
    //   lanes 16-31: row/col = lane - 16, K pair {k+2, k+3}
    const int mrow = lane & 15;
    const int kb   = (lane >> 4) * 2;
    const int mhi  = (lane >> 4) << 3;                  // +8 row offset for C/D

    for (int l = 0; l < 3; ++l) {
        __syncthreads();                                // Wt consumers done
        // Load layer-l weights transposed into LDS: Wt[n][k] = W[k][n]
        const float* Wg = W + l * (DIM * DIM);
        for (int i = tid; i < DIM * DIM; i += 256) {
            int k = i >> 6, n = i & 63;
            lds[W_OFF + n * WT_STRIDE + k] = Wg[i];
        }
        __syncthreads();                                // Wt ready

        // A fragments: 16 k-steps covering K=64
        v2f a[16];
        #pragma unroll
        for (int kk = 0; kk < 16; ++kk)
            a[kk] = *(const v2f*)(tile + mrow * TILE_STRIDE + kk * 4 + kb);

        // Accumulators (4 n-tiles), initialized with the bias (bias depends
        // only on N == lane%16, identical across the 8 C VGPRs).
        v8f acc[4];
        #pragma unroll
        for (int j = 0; j < 4; ++j) {
            float bb = lds[B_OFF + l * DIM + j * 16 + mrow];
            #pragma unroll
            for (int r = 0; r < 8; ++r) acc[j][r] = bb;
        }

        // 64 x V_WMMA_F32_16X16X4_F32 per layer per wave
        #pragma unroll
        for (int kk = 0; kk < 16; ++kk) {
            #pragma unroll
            for (int j = 0; j < 4; ++j) {
                v2f bf = *(const v2f*)(&lds[W_OFF + (j * 16 + mrow) * WT_STRIDE
                                            + kk * 4 + kb]);
                acc[j] = __builtin_amdgcn_wmma_f32_16x16x4_f32(
                    /*neg_a=*/false, a[kk], /*neg_b=*/false, bf,
                    /*c_mod=*/(short)0, acc[j],
                    /*reuse_a=*/false, /*reuse_b=*/false);
            }
        }

        if (l < 2) {
            // Activation; C layout -> row-major tile for the next layer's A.
            // lanes0-15 write rows r, lanes16-31 rows r+8: disjoint banks.
            #pragma unroll
            for (int j = 0; j < 4; ++j) {
                #pragma unroll
                for (int r = 0; r < 8; ++r) {
                    int m = r + mhi;
                    tile[m * TILE_STRIDE + j * 16 + mrow] = lrelu(acc[j][r]);
                }
            }
        } else {
            // Final layer: activation (+ optional residual fuse), store out.
            #pragma unroll
            for (int r = 0; r < 8; ++r) {
                int m = r + mhi;
                size_t row = (size_t)(node0 + m) * DIM;
                float mv = (mask != nullptr) ? mask[node0 + m] : 0.0f;
                #pragma unroll
                for (int j = 0; j < 4; ++j) {
                    float v = lrelu(acc[j][r]);
                    int col = j * 16 + mrow;
                    if (xres != nullptr) v = xres[row + col] + v * mv;
                    out[row + col] = v;
                }
            }
        }
    }
}

// Zero the segment-sum buffer (graph-capture safe; no memset API).
__global__ __launch_bounds__(256)
void zero_kernel(float4* __restrict__ p, int n4)
{
    int i = blockIdx.x * blockDim.x + threadIdx.x;
    if (i < n4) p[i] = make_float4(0.f, 0.f, 0.f, 0.f);
}

// s[dst] += y[src] over edges. 32 edges per wave: coalesced index loads,
// __shfl broadcast, each lane moves 2 floats of the 64-float row with a
// b64 load + two non-returning global_atomic_add_f32 (resolve in L2:
// both 32MB operand tensors fit in the 192MB L2).
__global__ __launch_bounds__(256)
void scatter_kernel(const float* __restrict__ y,
                    const int* __restrict__ dst,
                    const int* __restrict__ src,
                    float* __restrict__ s,
                    int nedges)
{
    int wave = (blockIdx.x * blockDim.x + threadIdx.x) >> 5;
    int lane = threadIdx.x & 31;
    int eb = wave * 32;
    if (eb >= nedges) return;
    int se = (eb + lane < nedges) ? src[eb + lane] : 0;
    int de = (eb + lane < nedges) ? dst[eb + lane] : 0;
    int cnt = (nedges - eb < 32) ? (nedges - eb) : 32;
    for (int i = 0; i < cnt; ++i) {
        int ss = __shfl(se, i);
        int dd = __shfl(de, i);
        float2 v = *(const float2*)(y + (size_t)ss * DIM + lane * 2);
        unsafeAtomicAdd(&s[(size_t)dd * DIM + lane * 2 + 0], v.x);
        unsafeAtomicAdd(&s[(size_t)dd * DIM + lane * 2 + 1], v.y);
    }
}

extern "C" void kernel_launch(void* const* d_in, const int* in_sizes, int n_in,
                              void* d_out, int out_size, void* d_ws, size_t ws_size,
                              hipStream_t stream)
{
    (void)in_sizes; (void)n_in; (void)out_size; (void)ws_size;

    const float* inputs = (const float*)d_in[0];                 // [N,64]
    const int*   edges  = (const int*)  d_in[1];                 // [8,2,E]
    const float* masks  = (const float*)d_in[2];                 // [8,N,1]
    const float* prep_W = (const float*)d_in[3];                 // [3,64,64]
    const float* prep_b = (const float*)d_in[4];                 // [3,64]
    const float* proc_W = (const float*)d_in[5];
    const float* proc_b = (const float*)d_in[6];
    const float* agg_W  = (const float*)d_in[7];
    const float* agg_b  = (const float*)d_in[8];

    const size_t ND = (size_t)N_NODES * DIM;                     // 8.39M floats
    float* X = (float*)d_ws;            // current node features
    float* Y = X + ND;                  // proc-MLP output
    float* S = Y + ND;                  // segment-sum accumulator

    const int mlp_blocks  = N_NODES / 128;                       // 1024
    const int zero_blocks = (int)(ND / 4 / 256);                 // 8192
    const int scat_blocks = (N_EDGES / 32 * 32) / 256;           // 8192

    // x = prep(inputs)
    mlp3_kernel<<<mlp_blocks, 256, 0, stream>>>(
        inputs, prep_W, prep_b, nullptr, nullptr, X);

    for (int d = 0; d < MAX_DEPTH; ++d) {
        // y = proc(x)
        mlp3_kernel<<<mlp_blocks, 256, 0, stream>>>(
            X, proc_W, proc_b, nullptr, nullptr, Y);

        // s = 0 ; s[dst] += y[src]
        zero_kernel<<<zero_blocks, 256, 0, stream>>>((float4*)S, (int)(ND / 4));
        const int* dst = edges + (size_t)d * 2 * N_EDGES;
        const int* src = dst + N_EDGES;
        scatter_kernel<<<scat_blocks, 256, 0, stream>>>(Y, dst, src, S, N_EDGES);

        // x = x + agg(s) * mask[d]   (last depth writes final output)
        float* outp = (d == MAX_DEPTH - 1) ? (float*)d_out : X;
        mlp3_kernel<<<mlp_blocks, 256, 0, stream>>>(
            S, agg_W, agg_b, X, masks + (size_t)d * N_NODES, outp);
    }
}